// Contrast_depth_loss_14843406975040
// MI455X (gfx1250) — compile-verified
//
#include <hip/hip_runtime.h>

// Contrast-depth loss: mean over (B, 8 neighbors, 30x30) of (d[n]-d[c])^2,
// where d = out - label. Pure bandwidth-bound streaming reduction:
// 134 MB in -> 1 float out. LDS-staged, WMMA-assisted wave reduction.

#define THREADS        256
#define WAVES          (THREADS / 32)
#define IMG_PER_BLOCK  8
#define RS             33          // padded LDS row stride (floats); gcd(33,64)=1 -> conflict-free
#define NBLOCKS        512

typedef __attribute__((ext_vector_type(2))) float v2f;
typedef __attribute__((ext_vector_type(8))) float v8f;

// Wave32 sum using V_WMMA_F32_16X16X4_F32.
// A (16x4, documented layout): lane L (L<16) supplies row L elements K=0,1;
// lane L+16 supplies row L elements K=2,3.  We put the partial in K-slot 0 of
// each lane and zero elsewhere; B = all ones (layout independent).
// => D[m][n] = p[m] + p[m+16] for all n.  D layout (documented): VGPR r holds
// M=r (lanes 0-15) / M=r+8 (lanes 16-31).  Summing a lane's 8 D regs gives
// rowsums 0..7 (low half-lanes) or 8..15 (high half-lanes); one shfl_xor(16)
// completes the 32-lane total (all lanes get it).
__device__ __forceinline__ float wave32_sum(float p) {
#if __has_builtin(__builtin_amdgcn_wmma_f32_16x16x4_f32)
    v2f a; a[0] = p;    a[1] = 0.0f;
    v2f b; b[0] = 1.0f; b[1] = 1.0f;
    v8f c = {0.f, 0.f, 0.f, 0.f, 0.f, 0.f, 0.f, 0.f};
    v8f d = __builtin_amdgcn_wmma_f32_16x16x4_f32(
        /*neg_a=*/false, a, /*neg_b=*/false, b,
        /*c_mod=*/(short)0, c, /*reuse_a=*/false, /*reuse_b=*/false);
    float s = ((d[0] + d[1]) + (d[2] + d[3])) + ((d[4] + d[5]) + (d[6] + d[7]));
    s += __shfl_xor(s, 16, 32);
    return s;
#else
    for (int off = 16; off > 0; off >>= 1) p += __shfl_xor(p, off, 32);
    return p;
#endif
}

__global__ __launch_bounds__(THREADS)
void cdl_main(const float* __restrict__ outp, const float* __restrict__ labp,
              float* __restrict__ partials, int numImages) {
    __shared__ float sd[IMG_PER_BLOCK * 32 * RS];
    __shared__ float wsum[WAVES];

    const int t = threadIdx.x;
    const int numGroups = (numImages + IMG_PER_BLOCK - 1) / IMG_PER_BLOCK;

    float acc = 0.0f;

    for (int g = blockIdx.x; g < numGroups; g += gridDim.x) {
        // Prefetch next group's data (one 128B line per thread per input).
        int gn = g + gridDim.x;
        if (gn < numGroups) {
            size_t nb = (size_t)gn * IMG_PER_BLOCK * 1024 + (size_t)t * 32;
            if (nb < (size_t)numImages * 1024) {
                __builtin_prefetch(outp + nb, 0, 0);
                __builtin_prefetch(labp + nb, 0, 0);
            }
        }

        __syncthreads();  // previous iteration's compute done before LDS overwrite

        // Stage d = out - label into LDS, coalesced float4 global loads.
        // 8 images * 1024 floats = 2048 float4s; 8 per thread.
#pragma unroll
        for (int v = 0; v < 8; ++v) {
            int q   = v * THREADS + t;   // float4 index within group [0,2048)
            int img = q >> 7;            // 128 float4 per 32x32 image
            int rem = q & 127;
            int row = rem >> 3;          // 8 float4 per row
            int c4  = rem & 7;
            size_t gimg = (size_t)g * IMG_PER_BLOCK + img;
            float4 o4 = make_float4(0.f, 0.f, 0.f, 0.f);
            float4 l4 = make_float4(0.f, 0.f, 0.f, 0.f);
            if (gimg < (size_t)numImages) {
                size_t off = gimg * 1024 + (size_t)row * 32 + (size_t)c4 * 4;
                o4 = *(const float4*)(outp + off);
                l4 = *(const float4*)(labp + off);
            }
            float* dst = &sd[(img * 32 + row) * RS + c4 * 4];
            dst[0] = o4.x - l4.x;
            dst[1] = o4.y - l4.y;
            dst[2] = o4.z - l4.z;
            dst[3] = o4.w - l4.w;
        }

        __syncthreads();

        // 240 interior rows (8 images x 30); thread t sweeps one row with a
        // 3x3 sliding register window: 3 LDS reads + 8 sub + 8 fma per pixel.
        if (t < IMG_PER_BLOCK * 30) {
            int img = t / 30;
            int i   = 1 + (t % 30);
            size_t gimg = (size_t)g * IMG_PER_BLOCK + img;
            if (gimg < (size_t)numImages) {
                const float* r0 = &sd[(img * 32 + (i - 1)) * RS];
                const float* r1 = r0 + RS;
                const float* r2 = r1 + RS;
                float a0 = r0[0], a1 = r0[1];
                float b0 = r1[0], b1 = r1[1];
                float c0 = r2[0], c1 = r2[1];
#pragma unroll
                for (int j = 1; j <= 30; ++j) {
                    float a2 = r0[j + 1];
                    float b2 = r1[j + 1];
                    float c2 = r2[j + 1];
                    float ctr = b1;
                    float e0 = a0 - ctr, e1 = a1 - ctr, e2 = a2 - ctr;
                    float e3 = b0 - ctr, e4 = b2 - ctr;
                    float e5 = c0 - ctr, e6 = c1 - ctr, e7 = c2 - ctr;
                    acc = fmaf(e0, e0, acc);
                    acc = fmaf(e1, e1, acc);
                    acc = fmaf(e2, e2, acc);
                    acc = fmaf(e3, e3, acc);
                    acc = fmaf(e4, e4, acc);
                    acc = fmaf(e5, e5, acc);
                    acc = fmaf(e6, e6, acc);
                    acc = fmaf(e7, e7, acc);
                    a0 = a1; a1 = a2;
                    b0 = b1; b1 = b2;
                    c0 = c1; c1 = c2;
                }
            }
        }
    }

    // Block reduction: WMMA wave reduce, then cross-wave via LDS.
    // (Converged here; EXEC all ones as WMMA requires.)
    float s = wave32_sum(acc);
    if ((t & 31) == 0) wsum[t >> 5] = s;
    __syncthreads();
    if (t == 0) {
        float tot = 0.f;
#pragma unroll
        for (int w = 0; w < WAVES; ++w) tot += wsum[w];
        partials[blockIdx.x] = tot;
    }
}

__global__ __launch_bounds__(THREADS)
void cdl_reduce(const float* __restrict__ partials, int n, float scale,
                float* __restrict__ result) {
    __shared__ float wsum[WAVES];
    float acc = 0.f;
    for (int i = threadIdx.x; i < n; i += THREADS) acc += partials[i];
    float s = wave32_sum(acc);
    if ((threadIdx.x & 31) == 0) wsum[threadIdx.x >> 5] = s;
    __syncthreads();
    if (threadIdx.x == 0) {
        float tot = 0.f;
#pragma unroll
        for (int w = 0; w < WAVES; ++w) tot += wsum[w];
        result[0] = tot * scale;
    }
}

extern "C" void kernel_launch(void* const* d_in, const int* in_sizes, int n_in,
                              void* d_out, int out_size, void* d_ws, size_t ws_size,
                              hipStream_t stream) {
    const float* outp = (const float*)d_in[0];
    const float* labp = (const float*)d_in[1];
    float* partials   = (float*)d_ws;

    const int numImages = in_sizes[0] / (32 * 32);   // B = 16384
    int numGroups = (numImages + IMG_PER_BLOCK - 1) / IMG_PER_BLOCK;
    int grid = NBLOCKS;
    if (grid > numGroups) grid = numGroups;
    if (grid < 1) grid = 1;

    cdl_main<<<grid, THREADS, 0, stream>>>(outp, labp, partials, numImages);

    const float scale =
        (float)(1.0 / ((double)numImages * 8.0 * 30.0 * 30.0));
    cdl_reduce<<<1, THREADS, 0, stream>>>(partials, grid, scale, (float*)d_out);
}